// GCN_27676769256197
// MI455X (gfx1250) — compile-verified
//
#include <hip/hip_runtime.h>
#include <hip/hip_bf16.h>

typedef float v2f __attribute__((ext_vector_type(2)));
typedef float v8f __attribute__((ext_vector_type(8)));

static constexpr int kNW  = 50000;
static constexpr int kND  = 20000;
static constexpr int kEWW = 500000;
static constexpr int kEWD = 300000;

// ---------------------------------------------------------------- utilities
__global__ __launch_bounds__(256) void k_zero(float* __restrict__ p, int n) {
  int t = blockIdx.x * 256 + threadIdx.x;
  if (t < n) p[t] = 0.0f;
}

__global__ __launch_bounds__(256) void k_deg(const int* __restrict__ idx,
                                             float* __restrict__ deg, int n) {
  int t = blockIdx.x * 256 + threadIdx.x;
  if (t < n) atomicAdd(&deg[idx[t]], 1.0f);
}

__global__ __launch_bounds__(256) void k_rsqrt(float* __restrict__ p, int n) {
  int t = blockIdx.x * 256 + threadIdx.x;
  if (t < n) p[t] = rsqrtf(fmaxf(p[t], 1.0f));
}

// ------------------------------------------------- H = diag(rs) * X * W (f32 WMMA)
// out_dim fixed at 128. block = 256 threads = 8 waves; wave w owns 16-col tile w.
// Each block computes 48 rows (3 M-tiles per wave) so each B fragment (from W)
// is loaded once per k-step and reused by 3 independent WMMAs.
template <int K>
__global__ __launch_bounds__(256) void k_gemm_wmma(
    const float* __restrict__ X, const float* __restrict__ rs,
    const float* __restrict__ W, float* __restrict__ H, int n_rows) {
  constexpr int LSTR = K + 4;   // stride: 16B-aligned row base, 16 distinct banks
  constexpr int K4   = K / 4;
  __shared__ float sx[48 * LSTR];
  const int tid  = threadIdx.x;
  const int row0 = blockIdx.x * 48;
  // stage 48 scaled rows into LDS as float4; clamp rows in the tail block
  // (48*K4 % 256 == 0 -> uniform trip count, EXEC stays all-ones)
  for (int idx = tid; idx < 48 * K4; idx += 256) {
    const int r  = idx / K4;
    const int c4 = idx - r * K4;
    int rr = row0 + r;
    rr = rr < n_rows ? rr : n_rows - 1;   // branchless clamp
    const float s = rs[rr];
    float4 v = *reinterpret_cast<const float4*>(X + (size_t)rr * K + c4 * 4);
    v.x *= s; v.y *= s; v.z *= s; v.w *= s;
    *reinterpret_cast<float4*>(&sx[r * LSTR + c4 * 4]) = v;
  }
  __syncthreads();
  const int wave = tid >> 5;
  const int lane = tid & 31;
  const int hi   = lane >> 4;                 // 0: K pair 0/1 ; 1: K pair 2/3
  const int l15  = lane & 15;
  const int col  = wave * 16 + l15;           // global output column
  v8f acc[3] = {{}, {}, {}};
  #pragma unroll 2
  for (int k = 0; k < K; k += 4) {
    const int ka = k + 2 * hi;
    v2f b;                                    // B: 4x16 tile, N = l15 (shared by 3 tiles)
    b.x = W[(size_t)ka * 128 + col];
    b.y = W[(size_t)(ka + 1) * 128 + col];
    #pragma unroll
    for (int mt = 0; mt < 3; ++mt) {
      v2f a;                                  // A: 16x4 tile, M = l15 (M-tile mt)
      a.x = sx[(mt * 16 + l15) * LSTR + ka];
      a.y = sx[(mt * 16 + l15) * LSTR + ka + 1];
      acc[mt] = __builtin_amdgcn_wmma_f32_16x16x4_f32(
          /*neg_a=*/false, a, /*neg_b=*/false, b,
          /*c_mod=*/(short)0, acc[mt], /*reuse_a=*/false, /*reuse_b=*/false);
    }
  }
  // D layout: VGPR v -> M = v + 8*hi, N = l15 ; guard tail rows at store only
  #pragma unroll
  for (int mt = 0; mt < 3; ++mt) {
    #pragma unroll
    for (int v = 0; v < 8; ++v) {
      const int row = row0 + mt * 16 + v + 8 * hi;
      if (row < n_rows) H[(size_t)row * 128 + col] = acc[mt][v];
    }
  }
}

// ---------------------------------------------- edge scatter: agg[dst] += h[src]*rs_i[dst]
__global__ __launch_bounds__(256) void k_scatter(
    const int* __restrict__ src, const int* __restrict__ dst,
    const float* __restrict__ rs_i, const float* __restrict__ h,
    float* __restrict__ agg, int n_edges) {
  int t = blockIdx.x * 256 + threadIdx.x;
  int e = t >> 5;                              // one wave per edge
  if (e >= n_edges) return;
  int lane = t & 31;                           // 4 columns per lane
  int s = src[e];
  int d = dst[e];
  float r = rs_i[d];
  const float4 v = *reinterpret_cast<const float4*>(h + (size_t)s * 128 + lane * 4);
  float* a = agg + (size_t)d * 128 + lane * 4;
  atomicAdd(a + 0, v.x * r);
  atomicAdd(a + 1, v.y * r);
  atomicAdd(a + 2, v.z * r);
  atomicAdd(a + 3, v.w * r);
}

__global__ __launch_bounds__(256) void k_bias_relu3(
    const float* __restrict__ agg, const float* __restrict__ b0,
    const float* __restrict__ b1, const float* __restrict__ b2,
    float* __restrict__ out, int n) {
  int t = blockIdx.x * 256 + threadIdx.x;
  if (t < n) {
    int j = t & 127;
    out[t] = fmaxf(agg[t] + b0[j] + b1[j] + b2[j], 0.0f);
  }
}

__global__ __launch_bounds__(256) void k_bias_relu1(
    const float* __restrict__ agg, const float* __restrict__ b0,
    float* __restrict__ out, int n) {
  int t = blockIdx.x * 256 + threadIdx.x;
  if (t < n) {
    int j = t & 127;
    out[t] = fmaxf(agg[t] + b0[j], 0.0f);
  }
}

// ---------------------------------------------- head: sigmoid(x @ lin_w + lin_b)
__global__ __launch_bounds__(256) void k_head(
    const float* __restrict__ x, const float* __restrict__ lw,
    const float* __restrict__ lb, float* __restrict__ out, int n_rows) {
  int t = blockIdx.x * 256 + threadIdx.x;
  int i = t >> 5;                              // one wave per node
  if (i >= n_rows) return;
  int lane = t & 31;
  const float4 v = *reinterpret_cast<const float4*>(x + (size_t)i * 128 + lane * 4);
  const float4 w = *reinterpret_cast<const float4*>(lw + lane * 4);
  float s = v.x * w.x + v.y * w.y + v.z * w.z + v.w * w.w;
  #pragma unroll
  for (int m = 16; m > 0; m >>= 1) s += __shfl_xor(s, m, 32);
  if (lane == 0) out[i] = 1.0f / (1.0f + expf(-(s + lb[0])));
}

// ----------------------------------------------------------------- launcher
extern "C" void kernel_launch(void* const* d_in, const int* in_sizes, int n_in,
                              void* d_out, int out_size, void* d_ws, size_t ws_size,
                              hipStream_t stream) {
  (void)in_sizes; (void)n_in; (void)out_size; (void)ws_size;
  const float* x_word = (const float*)d_in[0];
  const float* x_doc  = (const float*)d_in[1];
  const int* src_ww   = (const int*)d_in[2];
  const int* dst_ww   = (const int*)d_in[3];
  const int* src_wd   = (const int*)d_in[4];
  const int* dst_wd   = (const int*)d_in[5];
  const int* src_wwr  = (const int*)d_in[6];
  const int* dst_wwr  = (const int*)d_in[7];
  const int* src_dwr  = (const int*)d_in[8];
  const int* dst_dwr  = (const int*)d_in[9];
  const float* W1_ww  = (const float*)d_in[10]; const float* b1_ww  = (const float*)d_in[11];
  const float* W1_wd  = (const float*)d_in[12]; const float* b1_wd  = (const float*)d_in[13];
  const float* W1_wwr = (const float*)d_in[14]; const float* b1_wwr = (const float*)d_in[15];
  const float* W1_dwr = (const float*)d_in[16]; const float* b1_dwr = (const float*)d_in[17];
  const float* W2_ww  = (const float*)d_in[18]; const float* b2_ww  = (const float*)d_in[19];
  const float* W2_wd  = (const float*)d_in[20]; const float* b2_wd  = (const float*)d_in[21];
  const float* W2_wwr = (const float*)d_in[22]; const float* b2_wwr = (const float*)d_in[23];
  const float* W2_dwr = (const float*)d_in[24]; const float* b2_dwr = (const float*)d_in[25];
  const float* lin_w  = (const float*)d_in[26];
  const float* lin_b  = (const float*)d_in[27];
  float* out = (float*)d_out;

  // ---- carve workspace (float units, 1KB-aligned blocks) ----
  float* ws = (float*)d_ws;
  size_t off = 0;
  auto take = [&](size_t nf) {
    float* p = ws + off;
    off += (nf + 255) & ~(size_t)255;
    return p;
  };
  const size_t RS_TOTAL = 6 * (size_t)kNW + 2 * (size_t)kND;  // contiguous rs block
  float* rs_blk     = take(RS_TOTAL);
  float* rs_src_ww  = rs_blk;
  float* rs_dst_ww  = rs_src_ww  + kNW;
  float* rs_src_wd  = rs_dst_ww  + kNW;
  float* rs_dst_wd  = rs_src_wd  + kNW;
  float* rs_src_wwr = rs_dst_wd  + kND;
  float* rs_dst_wwr = rs_src_wwr + kNW;
  float* rs_src_dwr = rs_dst_wwr + kNW;
  float* rs_dst_dwr = rs_src_dwr + kND;
  float* h     = take((size_t)kNW * 128);       // reused per-relation GEMM output
  float* agg_w = take((size_t)kNW * 128);       // agg_w / agg_d contiguous
  float* agg_d = take((size_t)kND * 128);
  float* xw    = take((size_t)kNW * 128);
  float* xd    = take((size_t)kND * 128);

  auto cdiv = [](long long a, long long b) { return (int)((a + b - 1) / b); };
  const int NWf = kNW * 128, NDf = kND * 128;
  const int gW = cdiv(kNW, 48), gD = cdiv(kND, 48);   // GEMM grids (48 rows/block)

  // ---- degrees (shared by both layers) ----
  k_zero<<<cdiv((long long)RS_TOTAL, 256), 256, 0, stream>>>(rs_blk, (int)RS_TOTAL);
  k_deg<<<cdiv(kEWW, 256), 256, 0, stream>>>(src_ww,  rs_src_ww,  kEWW);
  k_deg<<<cdiv(kEWW, 256), 256, 0, stream>>>(dst_ww,  rs_dst_ww,  kEWW);
  k_deg<<<cdiv(kEWD, 256), 256, 0, stream>>>(src_wd,  rs_src_wd,  kEWD);
  k_deg<<<cdiv(kEWD, 256), 256, 0, stream>>>(dst_wd,  rs_dst_wd,  kEWD);
  k_deg<<<cdiv(kEWW, 256), 256, 0, stream>>>(src_wwr, rs_src_wwr, kEWW);
  k_deg<<<cdiv(kEWW, 256), 256, 0, stream>>>(dst_wwr, rs_dst_wwr, kEWW);
  k_deg<<<cdiv(kEWD, 256), 256, 0, stream>>>(src_dwr, rs_src_dwr, kEWD);
  k_deg<<<cdiv(kEWD, 256), 256, 0, stream>>>(dst_dwr, rs_dst_dwr, kEWD);
  k_rsqrt<<<cdiv((long long)RS_TOTAL, 256), 256, 0, stream>>>(rs_blk, (int)RS_TOTAL);

  // ---- layer 1 (K = 256) ----
  k_zero<<<cdiv(NWf + NDf, 256), 256, 0, stream>>>(agg_w, NWf + NDf);
  k_gemm_wmma<256><<<gW, 256, 0, stream>>>(x_word, rs_src_ww,  W1_ww,  h, kNW);
  k_scatter<<<cdiv((long long)kEWW * 32, 256), 256, 0, stream>>>(src_ww,  dst_ww,  rs_dst_ww,  h, agg_w, kEWW);
  k_gemm_wmma<256><<<gW, 256, 0, stream>>>(x_word, rs_src_wwr, W1_wwr, h, kNW);
  k_scatter<<<cdiv((long long)kEWW * 32, 256), 256, 0, stream>>>(src_wwr, dst_wwr, rs_dst_wwr, h, agg_w, kEWW);
  k_gemm_wmma<256><<<gD, 256, 0, stream>>>(x_doc,  rs_src_dwr, W1_dwr, h, kND);
  k_scatter<<<cdiv((long long)kEWD * 32, 256), 256, 0, stream>>>(src_dwr, dst_dwr, rs_dst_dwr, h, agg_w, kEWD);
  k_gemm_wmma<256><<<gW, 256, 0, stream>>>(x_word, rs_src_wd,  W1_wd,  h, kNW);
  k_scatter<<<cdiv((long long)kEWD * 32, 256), 256, 0, stream>>>(src_wd,  dst_wd,  rs_dst_wd,  h, agg_d, kEWD);
  k_bias_relu3<<<cdiv(NWf, 256), 256, 0, stream>>>(agg_w, b1_ww, b1_wwr, b1_dwr, xw, NWf);
  k_bias_relu1<<<cdiv(NDf, 256), 256, 0, stream>>>(agg_d, b1_wd, xd, NDf);

  // ---- layer 2 (K = 128) ----
  k_zero<<<cdiv(NWf + NDf, 256), 256, 0, stream>>>(agg_w, NWf + NDf);
  k_gemm_wmma<128><<<gW, 256, 0, stream>>>(xw, rs_src_ww,  W2_ww,  h, kNW);
  k_scatter<<<cdiv((long long)kEWW * 32, 256), 256, 0, stream>>>(src_ww,  dst_ww,  rs_dst_ww,  h, agg_w, kEWW);
  k_gemm_wmma<128><<<gW, 256, 0, stream>>>(xw, rs_src_wwr, W2_wwr, h, kNW);
  k_scatter<<<cdiv((long long)kEWW * 32, 256), 256, 0, stream>>>(src_wwr, dst_wwr, rs_dst_wwr, h, agg_w, kEWW);
  k_gemm_wmma<128><<<gD, 256, 0, stream>>>(xd, rs_src_dwr, W2_dwr, h, kND);
  k_scatter<<<cdiv((long long)kEWD * 32, 256), 256, 0, stream>>>(src_dwr, dst_dwr, rs_dst_dwr, h, agg_w, kEWD);
  k_gemm_wmma<128><<<gW, 256, 0, stream>>>(xw, rs_src_wd,  W2_wd,  h, kNW);
  k_scatter<<<cdiv((long long)kEWD * 32, 256), 256, 0, stream>>>(src_wd,  dst_wd,  rs_dst_wd,  h, agg_d, kEWD);
  k_bias_relu3<<<cdiv(NWf, 256), 256, 0, stream>>>(agg_w, b2_ww, b2_wwr, b2_dwr, xw, NWf);
  k_bias_relu1<<<cdiv(NDf, 256), 256, 0, stream>>>(agg_d, b2_wd, xd, NDf);

  // ---- head ----
  k_head<<<cdiv((long long)kNW * 32, 256), 256, 0, stream>>>(xw, lin_w, lin_b, out, kNW);
  k_head<<<cdiv((long long)kND * 32, 256), 256, 0, stream>>>(xd, lin_w, lin_b, out + kNW, kND);
}